// Model_84859963834701
// MI455X (gfx1250) — compile-verified
//
#include <hip/hip_runtime.h>

#define STEPS 8
#define VTH   0.5f
#define TAU   0.25f
#define EPSC  1e-5f

typedef __attribute__((ext_vector_type(2))) float v2f;
typedef __attribute__((ext_vector_type(4))) float v4f;
typedef __attribute__((ext_vector_type(8))) float v8f;

#define NBLK_RED 512
// ws layout (floats): [0, 2560) block partials (512 x 5); [2560, 2592) folded params
#define P_OFF 2560
// params p[]: [0..7]=W1' (o-major, o*2+c), [8..11]=b1', [12..19]=W2'/8 (o*4+c), [20..21]=b2

// ---------------- Pass 1: 5 raw moments of x ----------------
// S0=sum x0, S1=sum x1, Q00=sum x0^2, Q01=sum x0*x1, Q11=sum x1^2
__global__ __launch_bounds__(256) void snn_moments(const float* __restrict__ x,
                                                   float* __restrict__ partials,
                                                   int B) {
  float S0 = 0.f, S1 = 0.f, Q00 = 0.f, Q01 = 0.f, Q11 = 0.f;
  int stride = gridDim.x * blockDim.x;
  for (int i = blockIdx.x * blockDim.x + threadIdx.x; i < B; i += stride) {
    const float* xp = x + (size_t)i * 16;
    v4f a = *(const v4f*)(xp + 0);   // c0 t0..3
    v4f b = *(const v4f*)(xp + 4);   // c0 t4..7
    v4f c = *(const v4f*)(xp + 8);   // c1 t0..3
    v4f d = *(const v4f*)(xp + 12);  // c1 t4..7
#pragma unroll
    for (int j = 0; j < 4; ++j) {
      S0  += a[j] + b[j];
      S1  += c[j] + d[j];
      Q00 += a[j]*a[j] + b[j]*b[j];
      Q01 += a[j]*c[j] + b[j]*d[j];
      Q11 += c[j]*c[j] + d[j]*d[j];
    }
  }
  // wave32 butterfly reduce (ds_swizzle/permute path)
#pragma unroll
  for (int m = 16; m >= 1; m >>= 1) {
    S0  += __shfl_xor(S0, m);
    S1  += __shfl_xor(S1, m);
    Q00 += __shfl_xor(Q00, m);
    Q01 += __shfl_xor(Q01, m);
    Q11 += __shfl_xor(Q11, m);
  }
  __shared__ float wred[8][5];
  int wave = threadIdx.x >> 5;
  if ((threadIdx.x & 31) == 0) {
    wred[wave][0] = S0;  wred[wave][1] = S1;  wred[wave][2] = Q00;
    wred[wave][3] = Q01; wred[wave][4] = Q11;
  }
  __syncthreads();
  if (threadIdx.x == 0) {          // fixed-order -> deterministic
    float t[5] = {0.f, 0.f, 0.f, 0.f, 0.f};
    for (int w = 0; w < 8; ++w)
      for (int j = 0; j < 5; ++j) t[j] += wred[w][j];
    float* o = partials + (size_t)blockIdx.x * 5;
    for (int j = 0; j < 5; ++j) o[j] = t[j];
  }
}

// ---------------- Pass 1b: reduce partials, fold BN into layer params ----------------
__global__ __launch_bounds__(256) void snn_finalize(
    const float* __restrict__ partials, int nPart,
    const float* __restrict__ W1, const float* __restrict__ b1,
    const float* __restrict__ gamma, const float* __restrict__ beta,
    const float* __restrict__ W2, const float* __restrict__ b2,
    float* __restrict__ p, float invN) {
  __shared__ float sh[256];
  __shared__ float mom[5];
  for (int j = 0; j < 5; ++j) {
    float v = 0.f;
    for (int i = threadIdx.x; i < nPart; i += 256) v += partials[(size_t)i * 5 + j];
    sh[threadIdx.x] = v;
    __syncthreads();
    for (int s = 128; s > 0; s >>= 1) {  // fixed tree -> deterministic
      if (threadIdx.x < s) sh[threadIdx.x] += sh[threadIdx.x + s];
      __syncthreads();
    }
    if (threadIdx.x == 0) mom[j] = sh[0];
    __syncthreads();
  }
  if (threadIdx.x == 0) {
    float S0 = mom[0], S1 = mom[1], Q00 = mom[2], Q01 = mom[3], Q11 = mom[4];
    for (int o = 0; o < 4; ++o) {
      float w0 = W1[o*2+0], w1 = W1[o*2+1], bb = b1[o];
      // h = w0*x0 + w1*x1 + bb ; closed-form mean/var from raw moments of x
      float lin   = w0*S0 + w1*S1;
      float mean  = lin * invN + bb;
      float Eh2   = (w0*w0*Q00 + 2.f*w0*w1*Q01 + w1*w1*Q11) * invN
                    + 2.f*bb*lin*invN + bb*bb;
      float var   = Eh2 - mean*mean;
      float scale = gamma[o] * rsqrtf(var + EPSC);
      float shift = beta[o] - mean * scale;
      p[o*2+0] = w0 * scale;           // W1'
      p[o*2+1] = w1 * scale;
      p[8+o]   = bb * scale + shift;   // b1'
    }
    for (int j = 0; j < 2; ++j) {
      for (int c = 0; c < 4; ++c) p[12 + j*4 + c] = W2[j*4 + c] * (1.f / STEPS);
      p[20 + j] = b2[j];
    }
  }
}

// ---------------- Pass 2: WMMA layer1 + LIF + decode ----------------
// V_WMMA_F32_16X16X4_F32 with weights-in-A trick:
//   A rows 0-3 = W1' at k=0,1 (lanes 0-3);  A rows 8-11 = W1' at k=2,3 (lanes 24-27)
//   B rows k=0,1 live in lanes 0-15, k=2,3 in lanes 16-31 -> every lane supplies its
//   OWN sample's {x0[t], x1[t]} pair; A's zeros annihilate the mismatched halves.
//   D: lane L<16 gets rows 0-3 (its h[0..3]); lane L>=16 gets rows 8-11 (its h[0..3]).
//   C broadcasts b1' -> bias is free. Zero cross-lane data movement.
__global__ __launch_bounds__(256) void snn_main(const float* __restrict__ x,
                                                const float* __restrict__ p,
                                                float* __restrict__ out, int B) {
  int tid = blockIdx.x * 256 + threadIdx.x;
  int b = tid < B ? tid : B - 1;   // clamp via select: EXEC stays all-ones for WMMA
  const float* xp = x + (size_t)b * 16;
  v4f xa = *(const v4f*)(xp + 0);
  v4f xb = *(const v4f*)(xp + 4);
  v4f xc = *(const v4f*)(xp + 8);
  v4f xd = *(const v4f*)(xp + 12);
  float xc0[8] = {xa[0], xa[1], xa[2], xa[3], xb[0], xb[1], xb[2], xb[3]};
  float xc1[8] = {xc[0], xc[1], xc[2], xc[3], xd[0], xd[1], xd[2], xd[3]};

  float w1p[8], b1p[4], w2p[8], b2p[2];
#pragma unroll
  for (int i = 0; i < 8; ++i) w1p[i] = p[i];
#pragma unroll
  for (int i = 0; i < 4; ++i) b1p[i] = p[8 + i];
#pragma unroll
  for (int i = 0; i < 8; ++i) w2p[i] = p[12 + i];
#pragma unroll
  for (int i = 0; i < 2; ++i) b2p[i] = p[20 + i];

  float u0 = 0.f, u1 = 0.f, u2 = 0.f, u3 = 0.f;
  float s0 = 0.f, s1 = 0.f, s2 = 0.f, s3 = 0.f;
  float n0 = 0.f, n1 = 0.f, n2 = 0.f, n3 = 0.f;

#if defined(__AMDGCN__) && __has_builtin(__builtin_amdgcn_wmma_f32_16x16x4_f32)
  int lane = threadIdx.x & 31;
  int r = lane & 3;
  bool act = (lane < 4) || (lane >= 24 && lane < 28);
  float a0 = r == 0 ? w1p[0] : r == 1 ? w1p[2] : r == 2 ? w1p[4] : w1p[6];
  float a1 = r == 0 ? w1p[1] : r == 1 ? w1p[3] : r == 2 ? w1p[5] : w1p[7];
  v2f Am; Am[0] = act ? a0 : 0.f; Am[1] = act ? a1 : 0.f;
  v8f Cb = {b1p[0], b1p[1], b1p[2], b1p[3], 0.f, 0.f, 0.f, 0.f};
#pragma unroll
  for (int t = 0; t < STEPS; ++t) {
    v2f Bv; Bv[0] = xc0[t]; Bv[1] = xc1[t];
    v8f d = __builtin_amdgcn_wmma_f32_16x16x4_f32(
        false, Am, false, Bv, (short)0, Cb, false, false);
    float I0 = d[0], I1 = d[1], I2 = d[2], I3 = d[3];
    u0 = TAU * u0 * (1.f - s0) + I0; s0 = u0 > VTH ? 1.f : 0.f; n0 += s0;
    u1 = TAU * u1 * (1.f - s1) + I1; s1 = u1 > VTH ? 1.f : 0.f; n1 += s1;
    u2 = TAU * u2 * (1.f - s2) + I2; s2 = u2 > VTH ? 1.f : 0.f; n2 += s2;
    u3 = TAU * u3 * (1.f - s3) + I3; s3 = u3 > VTH ? 1.f : 0.f; n3 += s3;
  }
#else
#pragma unroll
  for (int t = 0; t < STEPS; ++t) {
    float I0 = w1p[0]*xc0[t] + w1p[1]*xc1[t] + b1p[0];
    float I1 = w1p[2]*xc0[t] + w1p[3]*xc1[t] + b1p[1];
    float I2 = w1p[4]*xc0[t] + w1p[5]*xc1[t] + b1p[2];
    float I3 = w1p[6]*xc0[t] + w1p[7]*xc1[t] + b1p[3];
    u0 = TAU * u0 * (1.f - s0) + I0; s0 = u0 > VTH ? 1.f : 0.f; n0 += s0;
    u1 = TAU * u1 * (1.f - s1) + I1; s1 = u1 > VTH ? 1.f : 0.f; n1 += s1;
    u2 = TAU * u2 * (1.f - s2) + I2; s2 = u2 > VTH ? 1.f : 0.f; n2 += s2;
    u3 = TAU * u3 * (1.f - s3) + I3; s3 = u3 > VTH ? 1.f : 0.f; n3 += s3;
  }
#endif
  // rate decode commuted with W2: out = sum_c count[c]*W2[o,c]/8 + b2[o]
  float o0 = n0*w2p[0] + n1*w2p[1] + n2*w2p[2] + n3*w2p[3] + b2p[0];
  float o1 = n0*w2p[4] + n1*w2p[5] + n2*w2p[6] + n3*w2p[7] + b2p[1];
  v2f res; res[0] = o0; res[1] = o1;
  *(v2f*)(out + (size_t)b * 2) = res;
}

extern "C" void kernel_launch(void* const* d_in, const int* in_sizes, int n_in,
                              void* d_out, int out_size, void* d_ws, size_t ws_size,
                              hipStream_t stream) {
  const float* x     = (const float*)d_in[0];
  const float* W1    = (const float*)d_in[1];
  const float* b1    = (const float*)d_in[2];
  const float* gamma = (const float*)d_in[3];
  const float* beta  = (const float*)d_in[4];
  const float* W2    = (const float*)d_in[5];
  const float* b2    = (const float*)d_in[6];
  float* out = (float*)d_out;

  int B = in_sizes[0] / 16;  // x: [B, 2, 8]
  float* partials = (float*)d_ws;
  float* p        = (float*)d_ws + P_OFF;

  snn_moments<<<NBLK_RED, 256, 0, stream>>>(x, partials, B);
  snn_finalize<<<1, 256, 0, stream>>>(partials, NBLK_RED, W1, b1, gamma, beta,
                                      W2, b2, p, 1.0f / (8.0f * (float)B));
  int nb = (B + 255) / 256;
  snn_main<<<nb, 256, 0, stream>>>(x, p, out, B);
}